// MultistrikePooling_19378892440078
// MI455X (gfx1250) — compile-verified
//
#include <hip/hip_runtime.h>

// Multistrike pooling, 8192 x 8192 fp32 -> 8192 x 8192 fp32.
// Pure bandwidth problem: 512 MB total traffic -> ~22 us floor at 23.3 TB/s.
// Strategy: one workgroup per row. Stage the row into LDS with CDNA5
// GLOBAL_LOAD_ASYNC_TO_LDS_B128 (ASYNCcnt), build the b4 partial-sum array in
// LDS so the overlapping k=16 window is served from LDS, and emit one
// coalesced b128 store per output group.

#define BA_DIM   8192
#define DE_DIM   8192
#define NGROUP   (DE_DIM / 4)      // 2048 groups of 4 output columns per row
#define THREADS  256
#define GPT      (NGROUP / THREADS) // 8 groups per thread

__global__ __launch_bounds__(THREADS)
void multistrike_pool_kernel(const float* __restrict__ x, float* __restrict__ out)
{
    __shared__ float xs[DE_DIM];   // 32 KB: staged input row
    __shared__ float b4s[NGROUP];  //  8 KB: block-of-4 sums

    const int row = blockIdx.x;
    const int t   = threadIdx.x;
    const float* __restrict__ xrow = x + (size_t)row * DE_DIM;

    // ---- Phase 1: async DMA the whole row into LDS (CDNA5 async path) ----
    // Each lane moves 16 B per instruction; consecutive lanes hit consecutive
    // 16 B chunks -> fully coalesced. 8 instructions per wave, ASYNCcnt <= 8.
    #pragma unroll
    for (int i = 0; i < GPT; ++i) {
        const int elem = 4 * (i * THREADS + t);              // float index
        unsigned lds_addr = (unsigned)(size_t)(&xs[elem]);   // low 32b = LDS byte addr
        const float* gptr = xrow + elem;
        asm volatile("global_load_async_to_lds_b128 %0, %1, off"
                     :: "v"(lds_addr), "v"(gptr)
                     : "memory");
    }
    // Drain this wave's async transfers, then barrier so every wave's LDS
    // writes are visible to the whole workgroup.
    asm volatile("s_wait_asynccnt 0" ::: "memory");
    __syncthreads();

    // ---- Phase 2: per-group 4-sums + first three interleaved outputs ----
    float o0[GPT], o1[GPT], o2[GPT];
    #pragma unroll
    for (int i = 0; i < GPT; ++i) {
        const int g = i * THREADS + t;
        const float4 a = reinterpret_cast<const float4*>(xs)[g]; // aligned ds_load_b128
        const float e = a.x + a.y;
        const float f = a.z + a.w;
        const float s = e + f;              // b4[g]
        o0[i] = e * 0.5f;                   // p2 even
        o1[i] = s * 0.25f;                  // p4
        o2[i] = f * 0.5f;                   // p2 odd
        b4s[g] = s;
    }
    __syncthreads();

    // ---- Phase 3: sliding k=16 pool from LDS b4 + coalesced b128 store ----
    float* __restrict__ orow = out + (size_t)row * DE_DIM;
    #pragma unroll
    for (int i = 0; i < GPT; ++i) {
        const int g = i * THREADS + t;
        // boundary fix: last 3 groups reuse the window 4 groups back
        const int h = (g < NGROUP - 3) ? g : (g - 4);
        const float w = b4s[h] + b4s[h + 1] + b4s[h + 2] + b4s[h + 3];
        float4 o;
        o.x = o0[i];
        o.y = o1[i];
        o.z = o2[i];
        o.w = w * 0.0625f;                  // f2 (k=16 pool / tail copy)
        reinterpret_cast<float4*>(orow)[g] = o;   // global_store_b128
    }
}

extern "C" void kernel_launch(void* const* d_in, const int* in_sizes, int n_in,
                              void* d_out, int out_size, void* d_ws, size_t ws_size,
                              hipStream_t stream)
{
    const float* x = (const float*)d_in[0];
    float* out     = (float*)d_out;
    (void)in_sizes; (void)n_in; (void)out_size; (void)d_ws; (void)ws_size;

    dim3 grid(BA_DIM);
    dim3 block(THREADS);
    hipLaunchKernelGGL(multistrike_pool_kernel, grid, block, 0, stream, x, out);
}